// TemporalPE_39616778338665
// MI455X (gfx1250) — compile-verified
//
#include <hip/hip_runtime.h>

#define B_ 8
#define L_ 2048
#define D_ 256
#define WAVES 4
#define NTHREADS (32 * WAVES)
#define JT (L_ / 16)       // 128 column tiles
#define KSTEPS (D_ / 4)    // 64 K-steps of 4 for f32 WMMA
#define LDSROW 260         // padded row stride (dwords) -> conflict-free b64 reads

typedef __attribute__((ext_vector_type(2))) float v2f;
typedef __attribute__((ext_vector_type(8))) float v8f;
typedef __attribute__((ext_vector_type(4))) int v4i;
typedef __attribute__((address_space(1))) v4i* gv4i_p;  // global int4*
typedef __attribute__((address_space(3))) v4i* lv4i_p;  // LDS int4*

#if __has_builtin(__builtin_amdgcn_global_load_async_to_lds_b128)
#define ASYNC_STAGE 1
#else
#define ASYNC_STAGE 0
#endif

// ---------------------------------------------------------------------------
// Kernel 1: sq[b*L + l] = ||x[b,l,:]||^2   (one wave per row)
// ---------------------------------------------------------------------------
__global__ __launch_bounds__(256) void TemporalPE_sq_kernel(
    const float* __restrict__ x, float* __restrict__ sq) {
  int warp = (blockIdx.x * 256 + threadIdx.x) >> 5;  // row index, exact fit
  int lane = threadIdx.x & 31;
  const float4* row = (const float4*)(x + (size_t)warp * D_);
  float4 a = row[lane];
  float4 c = row[lane + 32];
  float s = a.x * a.x + a.y * a.y + a.z * a.z + a.w * a.w +
            c.x * c.x + c.y * c.y + c.z * c.z + c.w * c.w;
#pragma unroll
  for (int off = 16; off; off >>= 1) s += __shfl_xor(s, off, 32);
  if (lane == 0) sq[warp] = s;
}

// ---------------------------------------------------------------------------
// Staging helpers: copy one 16x256 f32 j-tile into padded LDS.
// ---------------------------------------------------------------------------
#if ASYNC_STAGE
__device__ __forceinline__ void stage_tile_async(const float* gtile,
                                                 float* ldsbuf, int tid) {
  // 1024 float4s, 128 threads -> 8 async b128 copies per thread (512B/wave/op)
#pragma unroll
  for (int q = 0; q < 8; ++q) {
    int e = tid + NTHREADS * q;
    int row = e >> 6, c4 = e & 63;
    const float* g = gtile + row * D_ + c4 * 4;
    float* l = ldsbuf + row * LDSROW + c4 * 4;
    __builtin_amdgcn_global_load_async_to_lds_b128((gv4i_p)g, (lv4i_p)l, 0, 0);
  }
}
__device__ __forceinline__ void wait_async_done() {
#if __has_builtin(__builtin_amdgcn_s_wait_asynccnt)
  __builtin_amdgcn_s_wait_asynccnt(0);
#else
  asm volatile("s_wait_asynccnt 0x0" ::: "memory");
#endif
}
#else
__device__ __forceinline__ void stage_tile_sync(const float* gtile,
                                                float* ldsbuf, int tid) {
#pragma unroll
  for (int q = 0; q < 8; ++q) {
    int e = tid + NTHREADS * q;
    int row = e >> 6, c4 = e & 63;
    float4 v4 = ((const float4*)(gtile + (size_t)row * D_))[c4];
    int o = row * LDSROW + c4 * 4;
    ldsbuf[o + 0] = v4.x; ldsbuf[o + 1] = v4.y;
    ldsbuf[o + 2] = v4.z; ldsbuf[o + 3] = v4.w;
  }
}
#endif

// ---------------------------------------------------------------------------
// Kernel 2: fused Gram (f32 WMMA) + exp row-mean + (x + pe + w*mean)
// One WG = 4 waves, each wave owns a 16-row i-tile of one batch.
// B j-tiles double-buffered in LDS via async global->LDS copies.
// ---------------------------------------------------------------------------
__global__ __launch_bounds__(NTHREADS) void TemporalPE_main_kernel(
    const float* __restrict__ x, const float* __restrict__ pe,
    const float* __restrict__ wptr, const float* __restrict__ sq,
    float* __restrict__ out) {
  __shared__ float ldsB[2][16 * LDSROW];  // double-buffered shared j-tile
  __shared__ float ldsMean[WAVES * 16];   // per-wave row means

  const int tid  = threadIdx.x;
  const int wave = tid >> 5;
  const int lane = tid & 31;
  const int half = lane >> 4;  // 0: lanes 0-15, 1: lanes 16-31
  const int l16  = lane & 15;

  const int b     = blockIdx.x >> 5;        // 32 WGs per batch
  const int wgib  = blockIdx.x & 31;
  const int ibase = (wgib * WAVES + wave) * 16;

  const float* xb  = x + (size_t)b * L_ * D_;
  const float* sqb = sq + b * L_;

  // ---- Preload A tile (16 x 256) into registers, WMMA A-fragment layout:
  // lane l holds row (l&15); VGPR pair kk holds cols 4*kk + 2*half + {0,1}.
  v2f a_reg[KSTEPS];
  {
    const float* arow = xb + (size_t)(ibase + l16) * D_ + 2 * half;
#pragma unroll
    for (int kk = 0; kk < KSTEPS; ++kk)
      a_reg[kk] = *(const v2f*)(arow + 4 * kk);
  }

  // sq_i for the 8 rows covered by this lane's acc slots (m = v + 8*half)
  float sqi[8];
#pragma unroll
  for (int v = 0; v < 8; ++v) sqi[v] = sqb[ibase + v + 8 * half];

  float partial[8];
#pragma unroll
  for (int v = 0; v < 8; ++v) partial[v] = 0.0f;

#if ASYNC_STAGE
  // Prologue: prefetch tile 0 into buffer 0.
  stage_tile_async(xb, ldsB[0], tid);
#endif

  for (int jt = 0; jt < JT; ++jt) {
    const int jbase = jt * 16;
    const int cur = jt & 1;
#if ASYNC_STAGE
    wait_async_done();   // this wave's copies for ldsB[cur] have landed
    __syncthreads();     // all waves' copies visible; other buffer idle
    if (jt + 1 < JT)     // prefetch next tile while we compute this one
      stage_tile_async(xb + (size_t)(jbase + 16) * D_, ldsB[cur ^ 1], tid);
#else
    __syncthreads();
    stage_tile_sync(xb + (size_t)jbase * D_, ldsB[cur], tid);
    __syncthreads();
#endif

    float sqj = sqb[jbase + l16];  // N = l&15 in the C/D layout
    const float* bbuf = ldsB[cur];

    // ---- 16x16 Gram tile: 64 chained f32 WMMAs over K = 256.
    v8f acc = {};
#pragma unroll
    for (int kk = 0; kk < KSTEPS; ++kk) {
      // B fragment: lane l holds col N=(l&15), K = 4*kk + 2*half + {0,1}
      v2f bfrag = *(const v2f*)&bbuf[l16 * LDSROW + 4 * kk + 2 * half];
      acc = __builtin_amdgcn_wmma_f32_16x16x4_f32(
          false, a_reg[kk], false, bfrag, (short)0, acc, false, false);
    }

    // ---- sim = exp(-0.5 * max(sq_i + sq_j - 2 g, 0)); accumulate row sums.
#pragma unroll
    for (int v = 0; v < 8; ++v) {
      float s = sqi[v] + sqj - 2.0f * acc[v];
      s = fmaxf(s, 0.0f);
      partial[v] += __expf(-0.5f * s);
    }
  }

  // Reduce each row partial across the 16 lanes of its half-wave.
#pragma unroll
  for (int v = 0; v < 8; ++v) {
#pragma unroll
    for (int off = 1; off < 16; off <<= 1)
      partial[v] += __shfl_xor(partial[v], off, 16);
  }
  // Lanes 0 and 16 publish the 16 row means for this wave.
  if (l16 == 0) {
#pragma unroll
    for (int v = 0; v < 8; ++v)
      ldsMean[wave * 16 + half * 8 + v] = partial[v] * (1.0f / L_);
  }
  // Same-wave LDS RAW: compiler inserts the dscnt wait.

  // ---- Fused epilogue: out = x + pe + w * mean (16 rows x 64 float4s).
  const float w = wptr[0];
  float* ob = out + (size_t)b * L_ * D_;
  for (int e = lane; e < 16 * 64; e += 32) {
    int row = e >> 6, c4 = e & 63;
    float m = ldsMean[wave * 16 + row];
    float4 xv = ((const float4*)(xb + (size_t)(ibase + row) * D_))[c4];
    float4 pv = ((const float4*)(pe + (size_t)(ibase + row) * D_))[c4];
    float4 o;
    o.x = xv.x + pv.x + w * m;
    o.y = xv.y + pv.y + w * m;
    o.z = xv.z + pv.z + w * m;
    o.w = xv.w + pv.w + w * m;
    ((float4*)(ob + (size_t)(ibase + row) * D_))[c4] = o;
  }
}

// ---------------------------------------------------------------------------
extern "C" void kernel_launch(void* const* d_in, const int* in_sizes, int n_in,
                              void* d_out, int out_size, void* d_ws, size_t ws_size,
                              hipStream_t stream) {
  (void)in_sizes; (void)n_in; (void)out_size; (void)ws_size;
  const float* x  = (const float*)d_in[0];  // [B,L,D] fp32
  const float* sw = (const float*)d_in[1];  // [1] fp32
  const float* pe = (const float*)d_in[2];  // [1,L,D] fp32
  float* out = (float*)d_out;               // [B,L,D] fp32
  float* sq  = (float*)d_ws;                // B*L floats scratch (64 KB)

  TemporalPE_sq_kernel<<<(B_ * L_) / 8, 256, 0, stream>>>(x, sq);
  TemporalPE_main_kernel<<<B_ * (L_ / 16) / WAVES, NTHREADS, 0, stream>>>(
      x, pe, sw, sq, out);
}